// Conv2d_Mine_23424751632447
// MI455X (gfx1250) — compile-verified
//
#include <hip/hip_runtime.h>

typedef __attribute__((ext_vector_type(2))) float v2f;
typedef __attribute__((ext_vector_type(8))) float v8f;

#define H 1024
#define W 1024
#define KS 11
#define PAD 5
#define KP 28                    // K = 26 padded to multiple of 4
#define TSZ (KS * KP * 16)       // Toeplitz bank: 11 x 28 x 16 = 4928 floats

#define PATCH_ROWS 26            // 16 output rows + 10 halo
#define PCOLS_LOAD 144           // cols loaded: from x0-8 .. x0+135 (16B aligned, mult of 4)
#define PSTRIDE 148              // LDS row stride (148 mod 64 = 20 -> conflict-free A reads)
#define XSHIFT 3                 // patch col = wv*16 + k + XSHIFT (col 0 == x0-8, need x0+k-5)

// One-time Toeplitz precompute into global scratch:
// tdst[dy][k][j] = w[dy][k-j] if 0 <= k-j < 11 else 0   (rows k=26,27 all zero)
__global__ __launch_bounds__(256)
void build_toeplitz(const float* __restrict__ w, float* __restrict__ tdst) {
  int t = blockIdx.x * 256 + threadIdx.x;
  if (t < TSZ) {
    int dy  = t / (KP * 16);
    int rem = t % (KP * 16);
    int k   = rem >> 4;
    int j   = rem & 15;
    int d   = k - j;
    tdst[t] = (d >= 0 && d < KS) ? w[dy * KS + d] : 0.0f;
  }
}

template <bool PRE>
__global__ __launch_bounds__(256)
void conv11x11_wmma_f32(const float* __restrict__ x,
                        const float* __restrict__ wt,   // PRE: toeplitz global; else raw weights
                        float* __restrict__ out) {
  __shared__ alignas(16) float patch[PATCH_ROWS * PSTRIDE]; // 15392 B
  __shared__ alignas(16) float tmat[TSZ];                   // 19712 B

  const int tid = threadIdx.x;
  const int n   = blockIdx.z;
  const int y0  = blockIdx.y * 16;
  const int x0  = blockIdx.x * 128;

  const float* __restrict__ xin = x + (size_t)n * H * W;

  // ---- Toeplitz bank into LDS ----
  if (PRE) {
    const float4* __restrict__ tg = (const float4*)wt;     // 16B-aligned (hipMalloc)
    for (int t = tid; t < TSZ / 4; t += 256)
      ((float4*)tmat)[t] = tg[t];
  } else {
    for (int t = tid; t < TSZ; t += 256) {
      int dy  = t / (KP * 16);
      int rem = t % (KP * 16);
      int k   = rem >> 4;
      int j   = rem & 15;
      int d   = k - j;
      tmat[t] = (d >= 0 && d < KS) ? wt[dy * KS + d] : 0.0f;
    }
  }

  // ---- Input patch into LDS (rows y0-5..y0+20, cols x0-8..x0+135) ----
  const bool interior = (y0 >= PAD) && (y0 + 15 + PAD < H) &&
                        (x0 >= 8)   && (x0 + PCOLS_LOAD - 8 <= W);
  if (interior) {
    // Fully in-bounds: aligned float4 copy. 26 rows x 36 float4s = 936 b128 loads.
    const float4* __restrict__ src =
        (const float4*)(xin + (size_t)(y0 - PAD) * W + (x0 - 8));
    for (int t = tid; t < PATCH_ROWS * (PCOLS_LOAD / 4); t += 256) {
      int r = t / (PCOLS_LOAD / 4);
      int c = t % (PCOLS_LOAD / 4);
      ((float4*)&patch[r * PSTRIDE])[c] = src[(size_t)r * (W / 4) + c];
    }
  } else {
    // Border blocks: scalar loads with zero padding outside the image.
    for (int t = tid; t < PATCH_ROWS * PCOLS_LOAD; t += 256) {
      int r  = t / PCOLS_LOAD;
      int c  = t % PCOLS_LOAD;
      int gr = y0 - PAD + r;
      int gc = x0 - 8 + c;
      float v = 0.0f;
      if (gr >= 0 && gr < H && gc >= 0 && gc < W) v = xin[(size_t)gr * W + gc];
      patch[r * PSTRIDE + c] = v;
    }
  }
  __syncthreads();

  const int wv   = tid >> 5;      // wave id 0..7 -> x sub-tile (16 cols each)
  const int lane = tid & 31;
  const int half = lane >> 4;     // 0: K base +0, rows 0-7 ; 1: K base +2, rows 8-15
  const int l16  = lane & 15;

  v8f acc = {};

  for (int dy = 0; dy < KS; ++dy) {
    const float* __restrict__ prow = &patch[dy * PSTRIDE + wv * 16 + XSHIFT];
    const float* __restrict__ tdy  = &tmat[dy * (KP * 16)];
#pragma unroll
    for (int kc = 0; kc < KP / 4; ++kc) {
      const int kb = kc * 4 + half * 2;
      v2f a, b;
      // A (16x4 f32): lane m=l16; VGPR0 = K=kb, VGPR1 = K=kb+1 (halves offset by 2)
      a.x = prow[l16 * PSTRIDE + kb];
      a.y = prow[l16 * PSTRIDE + kb + 1];
      // B (4x16 f32): lane n=l16; VGPR0 = row kb, VGPR1 = row kb+1
      b.x = tdy[kb * 16 + l16];
      b.y = tdy[(kb + 1) * 16 + l16];
      acc = __builtin_amdgcn_wmma_f32_16x16x4_f32(
          /*neg_a=*/false, a, /*neg_b=*/false, b,
          /*c_mod=*/(short)0, acc, /*reuse_a=*/false, /*reuse_b=*/false);
    }
  }

  // C/D layout: VGPR r = (M=r, N=l16) for half 0, (M=r+8, N=l16) for half 1.
  float* __restrict__ obase =
      out + (size_t)n * H * W + (size_t)(y0 + half * 8) * W + (x0 + wv * 16 + l16);
#pragma unroll
  for (int r = 0; r < 8; ++r) obase[r * W] = acc[r];
}

extern "C" void kernel_launch(void* const* d_in, const int* in_sizes, int n_in,
                              void* d_out, int out_size, void* d_ws, size_t ws_size,
                              hipStream_t stream) {
  const float* x = (const float*)d_in[0];   // 64*1024*1024 f32
  const float* w = (const float*)d_in[1];   // 11*11 f32
  float* out = (float*)d_out;               // 64*1024*1024 f32
  (void)in_sizes; (void)n_in; (void)out_size;

  dim3 grid(W / 128, H / 16, 64);           // (8, 64, 64)
  dim3 block(256, 1, 1);                    // 8 wave32

  if (ws_size >= (size_t)TSZ * sizeof(float)) {
    float* tglob = (float*)d_ws;
    build_toeplitz<<<(TSZ + 255) / 256, 256, 0, stream>>>(w, tglob);
    conv11x11_wmma_f32<true><<<grid, block, 0, stream>>>(x, tglob, out);
  } else {
    conv11x11_wmma_f32<false><<<grid, block, 0, stream>>>(x, w, out);
  }
}